// GlobalAttention_32736240730712
// MI455X (gfx1250) — compile-verified
//
#include <hip/hip_runtime.h>
#include <hip/hip_bf16.h>

// Problem sizes (fixed by the reference).
constexpr int Bn = 8;
constexpr int Tn = 2048;
constexpr int Sn = 2048;
constexpr int Dn = 256;

typedef float v2f __attribute__((ext_vector_type(2)));
typedef float v8f __attribute__((ext_vector_type(8)));

__device__ __forceinline__ v8f wmma_f32_k4(v2f a, v2f b, v8f c) {
  // 8 args: (neg_a, A, neg_b, B, c_mod, C, reuse_a, reuse_b)
  return __builtin_amdgcn_wmma_f32_16x16x4_f32(
      false, a, false, b, (short)0, c, false, false);
}

// ---------------------------------------------------------------------------
// K1: align[b,t,s] = <source[b,t,:], memory[b,s,:]>  (raw scores, no mask)
// grid (S/512, T/16, B), block 256 (8 waves). Each wave: 16x16 output tile x4
// along s, sharing one A fragment per K-step.
// ---------------------------------------------------------------------------
__global__ void ga_scores_wmma(const float* __restrict__ src,
                               const float* __restrict__ mem,
                               float* __restrict__ alignv) {
  const int tid  = threadIdx.x;
  const int wave = tid >> 5;
  const int lane = tid & 31;
  const int b  = blockIdx.z;
  const int t0 = blockIdx.y * 16;
  const int s0 = blockIdx.x * 512 + wave * 64;  // this wave covers 4 tiles

  const int m    = lane & 15;
  const int koff = (lane >> 4) * 2;

  const float* aptr = src + ((size_t)(b * Tn + t0 + m)) * Dn + koff;
  const float* bptr0 = mem + ((size_t)(b * Sn + s0 +  0 + m)) * Dn + koff;
  const float* bptr1 = mem + ((size_t)(b * Sn + s0 + 16 + m)) * Dn + koff;
  const float* bptr2 = mem + ((size_t)(b * Sn + s0 + 32 + m)) * Dn + koff;
  const float* bptr3 = mem + ((size_t)(b * Sn + s0 + 48 + m)) * Dn + koff;

  v8f acc0 = {0.f,0.f,0.f,0.f,0.f,0.f,0.f,0.f};
  v8f acc1 = acc0, acc2 = acc0, acc3 = acc0;

#pragma unroll 4
  for (int k0 = 0; k0 < Dn; k0 += 4) {
    v2f af = *(const v2f*)(aptr + k0);
    v2f b0 = *(const v2f*)(bptr0 + k0);
    v2f b1 = *(const v2f*)(bptr1 + k0);
    v2f b2 = *(const v2f*)(bptr2 + k0);
    v2f b3 = *(const v2f*)(bptr3 + k0);
    acc0 = wmma_f32_k4(af, b0, acc0);
    acc1 = wmma_f32_k4(af, b1, acc1);
    acc2 = wmma_f32_k4(af, b2, acc2);
    acc3 = wmma_f32_k4(af, b3, acc3);
  }

  // C/D layout: VGPR v -> row (v + 8*(lane>>4)), col = lane&15
  const int rsel = (lane >> 4) * 8;
  const int col  = lane & 15;
  float* orow = alignv + (size_t)(b * Tn + t0 + rsel) * Sn + col;
#pragma unroll
  for (int v = 0; v < 8; ++v) {
    float* o = orow + (size_t)v * Sn;
    o[s0 +  0] = acc0[v];
    o[s0 + 16] = acc1[v];
    o[s0 + 32] = acc2[v];
    o[s0 + 48] = acc3[v];
  }
}

// ---------------------------------------------------------------------------
// K2: per (b,t) row: mask s>=len[b] to -inf, subtract row max, exp, in place.
// grid B*T blocks, 256 threads, 8 elems/thread.
// ---------------------------------------------------------------------------
__global__ void ga_mask_exp(float* __restrict__ alignv,
                            const int* __restrict__ lens) {
  const int row = blockIdx.x;          // b*T + t
  const int b   = row / Tn;
  const int tid = threadIdx.x;
  const int lane = tid & 31;
  const int wave = tid >> 5;
  const int len = lens[b];

  float* r = alignv + (size_t)row * Sn;
  float vals[8];
  float mx = -INFINITY;
#pragma unroll
  for (int j = 0; j < 8; ++j) {
    int s = tid + j * 256;
    float v = r[s];
    if (s >= len) v = -INFINITY;
    vals[j] = v;
    mx = fmaxf(mx, v);
  }
#pragma unroll
  for (int off = 16; off > 0; off >>= 1)
    mx = fmaxf(mx, __shfl_down(mx, off, 32));

  __shared__ float smax[8];
  if (lane == 0) smax[wave] = mx;
  __syncthreads();
  if (tid == 0) {
    float mm = smax[0];
#pragma unroll
    for (int j = 1; j < 8; ++j) mm = fmaxf(mm, smax[j]);
    smax[0] = mm;
  }
  __syncthreads();
  const float rowmax = smax[0];

#pragma unroll
  for (int j = 0; j < 8; ++j) {
    int s = tid + j * 256;
    r[s] = expf(vals[j] - rowmax);  // masked entries: exp(-inf) = 0
  }
}

// ---------------------------------------------------------------------------
// K3: sequential over t: u = E / (penalty + 1e-20), penalty = cumsum of E.
// One thread per (b,s). Also emits per-256-chunk partial row sums of u
// (deterministic, no atomics). grid (S/256, B), block 256.
// ---------------------------------------------------------------------------
__global__ void ga_cumsum(float* __restrict__ alignv,
                          float* __restrict__ partial) {
  const int b   = blockIdx.y;
  const int tid = threadIdx.x;
  const int lane = tid & 31;
  const int wave = tid >> 5;
  const int s = blockIdx.x * 256 + tid;

  float* base = alignv + (size_t)b * Tn * Sn + s;
  __shared__ float ssum[8];

  float run = 0.f;
  for (int t = 0; t < Tn; ++t) {
    float e = base[(size_t)t * Sn];
    float p = (t == 0) ? 1.f : run;
    float u = e / (p + 1e-20f);
    base[(size_t)t * Sn] = u;
    run += e;

    // block-wide sum of u
    float w = u;
#pragma unroll
    for (int off = 16; off > 0; off >>= 1)
      w += __shfl_down(w, off, 32);
    if (lane == 0) ssum[wave] = w;
    __syncthreads();
    if (tid == 0) {
      float tot = ssum[0];
#pragma unroll
      for (int j = 1; j < 8; ++j) tot += ssum[j];
      partial[((size_t)b * Tn + t) * 8 + blockIdx.x] = tot;
    }
    __syncthreads();
  }
}

// ---------------------------------------------------------------------------
// K4: normalize each row by its total (sum of the 8 chunk partials), in place.
// grid B*T blocks, 256 threads, 8 elems/thread.
// ---------------------------------------------------------------------------
__global__ void ga_rownorm(float* __restrict__ alignv,
                           const float* __restrict__ partial) {
  const int row = blockIdx.x;
  const int tid = threadIdx.x;
  float tot = 0.f;
#pragma unroll
  for (int j = 0; j < 8; ++j) tot += partial[(size_t)row * 8 + j];
  const float rinv = 1.f / tot;

  float* r = alignv + (size_t)row * Sn;
#pragma unroll
  for (int j = 0; j < 8; ++j) {
    int s = tid + j * 256;
    r[s] *= rinv;
  }
}

// ---------------------------------------------------------------------------
// K5: context GEMM  c[b,t,:] = align[b,t,:] @ memory[b,:,:]   (K = S = 2048)
// grid (D/128, T/32, B), block 256 (8 waves). Each wave: one 16-col d-tile,
// two t-tiles sharing the B fragment per K-step.
// ---------------------------------------------------------------------------
__global__ void ga_context_wmma(const float* __restrict__ alignv,
                                const float* __restrict__ mem,
                                float* __restrict__ cbuf) {
  const int tid  = threadIdx.x;
  const int wave = tid >> 5;
  const int lane = tid & 31;
  const int b  = blockIdx.z;
  const int t0 = blockIdx.y * 32;
  const int d0 = blockIdx.x * 128 + wave * 16;

  const int m  = lane & 15;   // A row within tile / B column within tile
  const int kb = (lane >> 4) * 2;

  const float* a0 = alignv + ((size_t)(b * Tn + t0 +      m)) * Sn + kb;
  const float* a1 = alignv + ((size_t)(b * Tn + t0 + 16 + m)) * Sn + kb;
  const float* mb = mem + (size_t)b * Sn * Dn + d0 + m;  // row s -> + s*Dn

  v8f acc0 = {0.f,0.f,0.f,0.f,0.f,0.f,0.f,0.f};
  v8f acc1 = acc0;

#pragma unroll 2
  for (int k0 = 0; k0 < Sn; k0 += 4) {
    const int kk = k0 + kb;
    v2f bf;
    bf.x = mb[(size_t)(kk    ) * Dn];
    bf.y = mb[(size_t)(kk + 1) * Dn];
    v2f af0 = *(const v2f*)(a0 + k0);
    v2f af1 = *(const v2f*)(a1 + k0);
    acc0 = wmma_f32_k4(af0, bf, acc0);
    acc1 = wmma_f32_k4(af1, bf, acc1);
  }

  const int rsel = (lane >> 4) * 8;
  const int col  = lane & 15;
#pragma unroll
  for (int v = 0; v < 8; ++v) {
    cbuf[(size_t)(b * Tn + t0 + v + rsel) * Dn + d0 + col]      = acc0[v];
    cbuf[(size_t)(b * Tn + t0 + 16 + v + rsel) * Dn + d0 + col] = acc1[v];
  }
}

// ---------------------------------------------------------------------------
// K6: attn_h = tanh([c | source] @ W_out), W_out is [2D, D] row-major.
// M = B*T rows, K = 512, N = 256. grid (D/128, M/16), block 256 (8 waves).
// ---------------------------------------------------------------------------
__global__ void ga_proj_wmma(const float* __restrict__ cbuf,
                             const float* __restrict__ src,
                             const float* __restrict__ W,
                             float* __restrict__ attn) {
  const int tid  = threadIdx.x;
  const int wave = tid >> 5;
  const int lane = tid & 31;
  const int row0 = blockIdx.y * 16;
  const int n0   = blockIdx.x * 128 + wave * 16;

  const int m  = lane & 15;
  const int kb = (lane >> 4) * 2;

  const float* crow = cbuf + (size_t)(row0 + m) * Dn;
  const float* srow = src  + (size_t)(row0 + m) * Dn;
  const float* wcol = W + n0 + (lane & 15);

  v8f acc = {0.f,0.f,0.f,0.f,0.f,0.f,0.f,0.f};

#pragma unroll 4
  for (int k0 = 0; k0 < 2 * Dn; k0 += 4) {
    const float* arow = (k0 < Dn) ? (crow + k0) : (srow + (k0 - Dn));
    v2f af = *(const v2f*)(arow + kb);
    const int kk = k0 + kb;
    v2f bf;
    bf.x = wcol[(size_t)(kk    ) * Dn];
    bf.y = wcol[(size_t)(kk + 1) * Dn];
    acc = wmma_f32_k4(af, bf, acc);
  }

  const int rsel = (lane >> 4) * 8;
  const int col  = lane & 15;
#pragma unroll
  for (int v = 0; v < 8; ++v)
    attn[(size_t)(row0 + v + rsel) * Dn + n0 + col] = tanhf(acc[v]);
}

// ---------------------------------------------------------------------------
// Launch: inputs = {source, memory_bank, W_out, memory_lengths};
// d_out = attn_h [B,T,D] then align_vectors [B,T,S] (align region doubles as
// scratch for scores/exp/u). Workspace: partial sums (B*T*8 f32) + c buffer.
// ---------------------------------------------------------------------------
extern "C" void kernel_launch(void* const* d_in, const int* in_sizes, int n_in,
                              void* d_out, int out_size, void* d_ws, size_t ws_size,
                              hipStream_t stream) {
  const float* src  = (const float*)d_in[0];
  const float* mem  = (const float*)d_in[1];
  const float* W    = (const float*)d_in[2];
  const int*   lens = (const int*)d_in[3];

  float* out    = (float*)d_out;
  float* attn   = out;                                // B*T*D
  float* alignv = out + (size_t)Bn * Tn * Dn;         // B*T*S

  float* partial = (float*)d_ws;                      // B*T*8 floats
  float* cbuf    = partial + (size_t)Bn * Tn * 8;     // B*T*D floats

  // 1) raw scores (WMMA f32)
  ga_scores_wmma<<<dim3(Sn / 512, Tn / 16, Bn), 256, 0, stream>>>(src, mem, alignv);
  // 2) mask + row max + exp
  ga_mask_exp<<<Bn * Tn, 256, 0, stream>>>(alignv, lens);
  // 3) coverage cumsum over t + partial row sums
  ga_cumsum<<<dim3(Sn / 256, Bn), 256, 0, stream>>>(alignv, partial);
  // 4) row normalization -> final align_vectors
  ga_rownorm<<<Bn * Tn, 256, 0, stream>>>(alignv, partial);
  // 5) context GEMM (WMMA f32)
  ga_context_wmma<<<dim3(Dn / 128, Tn / 32, Bn), 256, 0, stream>>>(alignv, mem, cbuf);
  // 6) output projection + tanh (WMMA f32)
  ga_proj_wmma<<<dim3(Dn / 128, (Bn * Tn) / 16), 256, 0, stream>>>(cbuf, src, W, attn);
}